// PARAFAC_45380624449891
// MI455X (gfx1250) — compile-verified
//
#include <hip/hip_runtime.h>

typedef float v2f __attribute__((ext_vector_type(2)));
typedef float v8f __attribute__((ext_vector_type(8)));

#define RANK 16
#define DIM  400
#define FACTOR_ELEMS (RANK * DIM)          // 6400 floats per factor
#define LDS_FLOATS   (3 * FACTOR_ELEMS)    // 19200 floats = 76.8 KB

// One wave handles 16 queries per tile iteration.
// D = A x B accumulated over four V_WMMA_F32_16X16X4_F32 ops:
//   A[q, r] = U[r, i_q] * V[r, j_q]     (16 x 16, queries x ranks)
//   B[r, q] = W[r, k_q]                 (16 x 16, ranks x queries)
//   out[q]  = D[q, q]
__global__ void parafac_query_wmma(const float* __restrict__ U,
                                   const float* __restrict__ V,
                                   const float* __restrict__ W,
                                   const int*   __restrict__ idx,
                                   float*       __restrict__ out,
                                   int nQueries)
{
    extern __shared__ float lds[];               // [U | V | W], 76.8 KB
    float* Ul = lds;
    float* Vl = lds + FACTOR_ELEMS;
    float* Wl = lds + 2 * FACTOR_ELEMS;

    // Cooperative stage of all three factor matrices into LDS.
    for (int t = threadIdx.x; t < FACTOR_ELEMS; t += blockDim.x) {
        Ul[t] = U[t];
        Vl[t] = V[t];
        Wl[t] = W[t];
    }
    __syncthreads();

    const int lane   = threadIdx.x & 31;
    const int q      = lane & 15;                // query slot within tile
    const int half   = lane >> 4;                // 0 -> K {0,1}, 1 -> K {2,3} (per 4-wide chunk)
    const int waveId = (blockIdx.x * blockDim.x + threadIdx.x) >> 5;
    const int nWaves = (gridDim.x * blockDim.x) >> 5;
    const int nTiles = (nQueries + 15) >> 4;

    for (int tile = waveId; tile < nTiles; tile += nWaves) {
        const int qi = tile * 16 + q;
        const int qc = (qi < nQueries) ? qi : (nQueries - 1);   // clamp; store is guarded

        // Coalesced index gather (lanes 0-15 and 16-31 fetch the same triples — cached).
        const int i = idx[qc * 3 + 0];
        const int j = idx[qc * 3 + 1];
        const int k = idx[qc * 3 + 2];

        // Prefetch next tile's index cacheline (gfx1250 global_prefetch_b8).
        {
            const int nq = (tile + nWaves) * 16 + q;
            if (nq < nQueries)
                __builtin_prefetch(&idx[nq * 3], 0, 1);
        }

        v8f c = {};
        #pragma unroll
        for (int t = 0; t < 4; ++t) {
            // Ranks this lane supplies for chunk t (A layout: 16x4 f32,
            // lanes 0-15 hold K=0,1 / lanes 16-31 hold K=2,3; B mirrors it).
            const int r0 = 4 * t + 2 * half;
            const int r1 = r0 + 1;

            v2f a, b;
            a.x = Ul[r0 * DIM + i] * Vl[r0 * DIM + j];
            a.y = Ul[r1 * DIM + i] * Vl[r1 * DIM + j];
            b.x = Wl[r0 * DIM + k];
            b.y = Wl[r1 * DIM + k];

            // 8 args: (neg_a, A, neg_b, B, c_mod, C, reuse_a, reuse_b)
            c = __builtin_amdgcn_wmma_f32_16x16x4_f32(
                    false, a, false, b, (short)0, c, false, false);
        }

        // Diagonal extraction per C/D layout:
        //   q in 0..7  -> lane q,      VGPR q
        //   q in 8..15 -> lane q + 16, VGPR q - 8
        if (lane < 8 || lane >= 24) {
            const int dq = (lane < 8) ? lane : (lane - 16);
            const int oq = tile * 16 + dq;
            if (oq < nQueries) {
                const int s = lane & 7;
                float vsel = c[0];
                if (s == 1) vsel = c[1];
                if (s == 2) vsel = c[2];
                if (s == 3) vsel = c[3];
                if (s == 4) vsel = c[4];
                if (s == 5) vsel = c[5];
                if (s == 6) vsel = c[6];
                if (s == 7) vsel = c[7];
                out[oq] = vsel;
            }
        }
    }
}

extern "C" void kernel_launch(void* const* d_in, const int* in_sizes, int n_in,
                              void* d_out, int out_size, void* d_ws, size_t ws_size,
                              hipStream_t stream) {
    const float* U   = (const float*)d_in[0];
    const float* V   = (const float*)d_in[1];
    const float* W   = (const float*)d_in[2];
    const int*   idx = (const int*)  d_in[3];
    float*       out = (float*)d_out;

    const int nQueries = out_size;                     // 100000
    const int nTiles   = (nQueries + 15) / 16;         // 16 queries per wave-tile

    const int threads       = 256;                     // 8 wave32s per block
    const int wavesPerBlock = threads / 32;
    int blocks = (nTiles + wavesPerBlock - 1) / wavesPerBlock;
    if (blocks > 128) blocks = 128;                    // amortize the 76.8 KB LDS fill
    if (blocks < 1)   blocks = 1;

    const size_t ldsBytes = (size_t)LDS_FLOATS * sizeof(float);

    hipLaunchKernelGGL(parafac_query_wmma, dim3(blocks), dim3(threads), ldsBytes, stream,
                       U, V, W, idx, out, nQueries);
}